// RotateHealEncoding_77764677862010
// MI455X (gfx1250) — compile-verified
//
#include <hip/hip_runtime.h>

typedef __attribute__((ext_vector_type(2))) float v2f;
typedef __attribute__((ext_vector_type(4))) float v4f;
typedef __attribute__((ext_vector_type(8))) float v8f;

#define N_LEVELS 4
#define FEAT 32
#define BATCH 262144
#define MAX_SIZE 792
#define WAVES_PER_BLOCK 8
#define THREADS (WAVES_PER_BLOCK * 32)

// out[l,b,f] = sum_{j=0..11} W[b,j] * params[l,j,f],
// W[b,j] = sum_k mask(l,k,b) * dist(l,k,b) * (neigh_index==j)  +  (pixel_index==j)
// Output element goes to out[b*128 + f*4 + l].
__global__ __launch_bounds__(THREADS) void heal_wmma_kernel(
    const float* __restrict__ params,        // [L, 792, 32]
    const float* __restrict__ pixel_latlon,  // [L, B, 2]
    const float* __restrict__ neigh_latlon,  // [L, 8*B, 2]
    const int*   __restrict__ pixel_index,   // [L, B]
    const int*   __restrict__ neigh_index,   // [L, 8, B]
    float* __restrict__ out)                 // [B, L*F] col = f*4 + l
{
  __shared__ float wsh[WAVES_PER_BLOCK][16][12];

  const int tid  = threadIdx.x;
  const int wave = tid >> 5;
  const int lane = tid & 31;
  const int row  = lane & 15;   // batch row within tile; also N (feature) for B/D
  const int half = lane >> 4;   // 0: lanes 0-15, 1: lanes 16-31
  const int tile = blockIdx.x * WAVES_PER_BLOCK + wave;
  const int base = tile * 16;
  const int b    = base + row;  // this lane's batch row

  v8f acc[N_LEVELS][2] = {};    // fp32 accumulators, [level][feature-tile]

  float* w0 = &wsh[wave][0][0];

  for (int l = 0; l < N_LEVELS; ++l) {
    // ---- zero the 16x12 weight block (192 floats, 6 per lane) ----
    #pragma unroll
    for (int i = lane; i < 16 * 12; i += 32) w0[i] = 0.0f;
    __syncthreads();

    // ---- weight accumulation: each lane handles 4 neighbors of its row ----
    const float myt = pixel_latlon[(size_t)(l * BATCH + b) * 2 + 0];
    const float myp = pixel_latlon[(size_t)(l * BATCH + b) * 2 + 1];
    #pragma unroll
    for (int kk = 0; kk < 4; ++kk) {
      const int k = half * 4 + kk;
      const size_t nidx = (size_t)(l * 8 + k) * BATCH + b;
      const int j = neigh_index[nidx];
      if (j >= 0) {
        const float nt = neigh_latlon[nidx * 2 + 0];
        const float np = neigh_latlon[nidx * 2 + 1];
        const float dt = nt - myt;
        const float dp = np - myp;
        const float dist = __builtin_sqrtf(dt * dt + dp * dp);
        atomicAdd(&wsh[wave][row][j], dist);   // ds_add_f32
      }
    }
    if (half == 0) {
      int pi = pixel_index[(size_t)l * BATCH + b];
      if (pi < 0) pi = 0;
      atomicAdd(&wsh[wave][row][pi], 1.0f);    // my_reps term
    }
    __syncthreads();

    // ---- WMMA: D(16x16 f32) += A(16x4) * B(4x16), K=12 in 3 chunks ----
    #pragma unroll
    for (int kc = 0; kc < 3; ++kc) {
      const int K = kc * 4 + half * 2;         // A layout: VGPR0 holds K / K+2 halves
      v2f a;
      a.x = wsh[wave][row][K];
      a.y = wsh[wave][row][K + 1];
      #pragma unroll
      for (int t = 0; t < 2; ++t) {
        v2f bm;                                 // B layout mirrored: lane = N = row
        bm.x = params[(size_t)(l * MAX_SIZE + K)     * FEAT + t * 16 + row];
        bm.y = params[(size_t)(l * MAX_SIZE + K + 1) * FEAT + t * 16 + row];
        acc[l][t] = __builtin_amdgcn_wmma_f32_16x16x4_f32(
            /*neg_a=*/false, a, /*neg_b=*/false, bm,
            /*c_mod=*/(short)0, acc[l][t],
            /*reuse_a=*/false, /*reuse_b=*/false);
      }
    }
    __syncthreads();  // protect LDS before next level's zeroing (cross-half reads)
  }

  // ---- store: D layout identical across levels -> pack float4 over l ----
  // D VGPR r: lanes0-15 -> M=r, lanes16-31 -> M=r+8; N = lane&15.
  #pragma unroll
  for (int r = 0; r < 8; ++r) {
    const int bb = base + r + half * 8;
    #pragma unroll
    for (int t = 0; t < 2; ++t) {
      const int f = t * 16 + row;
      v4f o;
      o.x = acc[0][t][r];
      o.y = acc[1][t][r];
      o.z = acc[2][t][r];
      o.w = acc[3][t][r];
      *(v4f*)(out + (size_t)bb * (N_LEVELS * FEAT) + f * N_LEVELS) = o;
    }
  }
}

extern "C" void kernel_launch(void* const* d_in, const int* in_sizes, int n_in,
                              void* d_out, int out_size, void* d_ws, size_t ws_size,
                              hipStream_t stream) {
  const float* params       = (const float*)d_in[0];
  const float* pixel_latlon = (const float*)d_in[1];
  const float* neigh_latlon = (const float*)d_in[2];
  const int*   pixel_index  = (const int*)d_in[3];
  const int*   neigh_index  = (const int*)d_in[4];
  float* out = (float*)d_out;

  const int tiles  = BATCH / 16;                 // 16384
  const int blocks = tiles / WAVES_PER_BLOCK;    // 2048
  heal_wmma_kernel<<<blocks, THREADS, 0, stream>>>(
      params, pixel_latlon, neigh_latlon, pixel_index, neigh_index, out);
}